// SteamNet_Autoencoder_64338610094308
// MI455X (gfx1250) — compile-verified
//
#include <hip/hip_runtime.h>
#include <math.h>

// ---------------------------------------------------------------------------
// Problem constants (from reference): B,N,W,F = 32,2000,24,12 ; H,G,E,L = 64,32,8000,3
// ---------------------------------------------------------------------------
constexpr int Bb = 32, Nn = 2000, Ww = 24, Ff = 12, Hh = 64, Gg = 32, Ee = 8000, Ll = 3;
constexpr int BT    = Bb * Ww;        // 768 (b,t) pairs
constexpr int ROWS  = Bb * Nn;        // 64000 GRU rows
constexpr int MT    = Nn / 16;        // 125 M-tiles per (b,t)
constexpr int TILES = BT * MT;        // 96000 wave-tiles
static_assert(Nn % 16 == 0, "N must be divisible by 16");
static_assert(ROWS % (16 * 8) == 0, "rows must fill blocks exactly");

typedef float v2f __attribute__((ext_vector_type(2)));
typedef float v8f __attribute__((ext_vector_type(8)));

// Exact-precision fp32 WMMA: D(16x16) = A(16x4) * B(4x16) + C
static __device__ __forceinline__ v8f wmma4(v2f a, v2f b, v8f c) {
  return __builtin_amdgcn_wmma_f32_16x16x4_f32(false, a, false, b, (short)0, c,
                                               false, false);
}
static __device__ __forceinline__ v8f splat8(float v) {
  v8f r;
#pragma unroll
  for (int i = 0; i < 8; ++i) r[i] = v;
  return r;
}
// Wave-synchronous LDS fence (each wave owns its own LDS slab -> no workgroup
// barrier needed). CDNA5 split counter: s_wait_dscnt.
static __device__ __forceinline__ void wave_lds_sync() {
  __builtin_amdgcn_wave_barrier();
  asm volatile("s_wait_dscnt 0" ::: "memory");
  __builtin_amdgcn_wave_barrier();
}
static __device__ __forceinline__ float sigmoidf_(float x) {
  return 1.0f / (1.0f + __expf(-x));
}

// ---------------------------------------------------------------------------
// GCN normalization precompute (PyG default: self loops + deg^-1/2 symmetric)
// ---------------------------------------------------------------------------
__global__ void k_deg_init(float* __restrict__ deg) {
  int i = blockIdx.x * blockDim.x + threadIdx.x;
  if (i < Nn) deg[i] = 1.0f;  // self loop
}
__global__ void k_deg_count(const int* __restrict__ dst, float* __restrict__ deg) {
  int e = blockIdx.x * blockDim.x + threadIdx.x;
  if (e < Ee) atomicAdd(&deg[dst[e]], 1.0f);
}
__global__ void k_rsqrt_inplace(float* __restrict__ deg) {
  int i = blockIdx.x * blockDim.x + threadIdx.x;
  if (i < Nn) deg[i] = rsqrtf(deg[i]);  // deg -> dis
}
__global__ void k_edgenorm(const int* __restrict__ src, const int* __restrict__ dst,
                           const float* __restrict__ dis, float* __restrict__ nw) {
  int e = blockIdx.x * blockDim.x + threadIdx.x;
  if (e < Ee) nw[e] = dis[src[e]] * dis[dst[e]];
}

// ---------------------------------------------------------------------------
// Projection: enc[b][t][n][h] = x[b][n][t][:12] @ proj_W + proj_b   (WMMA f32)
// one wave per 16-node tile; K=12 -> 3 k-steps; N=64 -> 4 n-tiles
// ---------------------------------------------------------------------------
__global__ __launch_bounds__(128) void k_project(const float* __restrict__ x,
                                                 const float* __restrict__ pW,
                                                 const float* __restrict__ pb,
                                                 float* __restrict__ enc) {
  __shared__ float sW[Ff * Hh];
  __shared__ float sb[Hh];
  for (int i = threadIdx.x; i < Ff * Hh; i += 128) sW[i] = pW[i];
  if (threadIdx.x < Hh) sb[threadIdx.x] = pb[threadIdx.x];
  __syncthreads();

  const int lane = threadIdx.x & 31, wave = threadIdx.x >> 5;
  const int lo = lane & 15, hi = lane >> 4;
  const int tile = blockIdx.x * 4 + wave;
  const int bt = tile / MT, mt = tile - bt * MT, n0 = mt * 16;
  const int b = bt / Ww, t = bt - b * Ww;

  v8f acc[4];
#pragma unroll
  for (int j = 0; j < 4; ++j) acc[j] = splat8(sb[16 * j + lo]);

#pragma unroll
  for (int k = 0; k < 3; ++k) {
    const int kk = 4 * k + 2 * hi;
    v2f a = *reinterpret_cast<const v2f*>(
        x + ((size_t)(b * Nn + n0 + lo) * Ww + t) * Ff + kk);
#pragma unroll
    for (int j = 0; j < 4; ++j) {
      v2f bb;
      bb.x = sW[kk * Hh + 16 * j + lo];
      bb.y = sW[(kk + 1) * Hh + 16 * j + lo];
      acc[j] = wmma4(a, bb, acc[j]);
    }
  }
  float* dstp = enc + (size_t)bt * Nn * Hh;
#pragma unroll
  for (int j = 0; j < 4; ++j)
#pragma unroll
    for (int p = 0; p < 8; ++p)
      dstp[(size_t)(n0 + p + 8 * hi) * Hh + 16 * j + lo] = acc[j][p];
}

// ---------------------------------------------------------------------------
// GCN matmul: out[bt][n][:] = in[bt][n][:] @ W_l  (64x64, WMMA f32, LDS-staged W)
// ---------------------------------------------------------------------------
__global__ __launch_bounds__(128) void k_gcn_mm(const float* __restrict__ in,
                                                const float* __restrict__ Wl,
                                                float* __restrict__ outp) {
  __shared__ float sW[Hh * Hh];
  for (int i = threadIdx.x; i < Hh * Hh; i += 128) sW[i] = Wl[i];
  __syncthreads();

  const int lane = threadIdx.x & 31, wave = threadIdx.x >> 5;
  const int lo = lane & 15, hi = lane >> 4;
  const int tile = blockIdx.x * 4 + wave;
  const int bt = tile / MT, mt = tile - bt * MT, n0 = mt * 16;
  const float* srcp = in + (size_t)bt * Nn * Hh;
  float* dstp = outp + (size_t)bt * Nn * Hh;

  v8f acc[4];
#pragma unroll
  for (int j = 0; j < 4; ++j) acc[j] = splat8(0.0f);

#pragma unroll
  for (int k = 0; k < 16; ++k) {
    const int kk = 4 * k + 2 * hi;
    v2f a = *reinterpret_cast<const v2f*>(srcp + (size_t)(n0 + lo) * Hh + kk);
#pragma unroll
    for (int j = 0; j < 4; ++j) {
      v2f bb;
      bb.x = sW[kk * Hh + 16 * j + lo];
      bb.y = sW[(kk + 1) * Hh + 16 * j + lo];
      acc[j] = wmma4(a, bb, acc[j]);
    }
  }
#pragma unroll
  for (int j = 0; j < 4; ++j)
#pragma unroll
    for (int p = 0; p < 8; ++p)
      dstp[(size_t)(n0 + p + 8 * hi) * Hh + 16 * j + lo] = acc[j][p];
}

// Self-loop term initializes the scatter target (non-atomic full write).
__global__ __launch_bounds__(256) void k_selfinit(const float* __restrict__ hw,
                                                  const float* __restrict__ dis,
                                                  float* __restrict__ outp) {
  int g = blockIdx.x * 256 + threadIdx.x;        // exactly BT*Nn*Hh threads
  int n = (g >> 6) % Nn;
  float d = dis[n];
  outp[g] = hw[g] * d * d;
}

// Edge scatter: out[bt][dst] += hw[bt][src] * norm_e  (f32 global atomics)
__global__ __launch_bounds__(256) void k_scatter(const float* __restrict__ hw,
                                                 const int* __restrict__ src,
                                                 const int* __restrict__ dst,
                                                 const float* __restrict__ nw,
                                                 float* __restrict__ outp) {
  int g = blockIdx.x * 256 + threadIdx.x;        // exactly BT*Ee*16 threads
  int c4 = g & 15;
  int te = g >> 4;
  int e = te % Ee;
  int bt = te / Ee;
  float w = nw[e];
  const float4 m = *reinterpret_cast<const float4*>(
      hw + ((size_t)bt * Nn + src[e]) * Hh + c4 * 4);
  float* o = outp + ((size_t)bt * Nn + dst[e]) * Hh + c4 * 4;
  atomicAdd(o + 0, m.x * w);
  atomicAdd(o + 1, m.y * w);
  atomicAdd(o + 2, m.z * w);
  atomicAdd(o + 3, m.w * w);
}

__global__ __launch_bounds__(256) void k_biasrelu(float* __restrict__ h,
                                                  const float* __restrict__ gb) {
  int g = blockIdx.x * 256 + threadIdx.x;
  int c = g & 63;
  h[g] = fmaxf(h[g] + gb[c], 0.0f);
}

// ---------------------------------------------------------------------------
// Fused 2-layer GRU + 6-head decoder. One wave owns 16 rows; hidden states are
// kept in registers (WMMA C layout) and mirrored into a per-wave LDS slab so
// they can be re-read as WMMA A fragments (the LDS acts as the transpose).
// r/z gates: the x@Wih accumulator is chained as C into the h@Whh WMMA, so
// only the sum (ir+hr, iz+hz) is ever materialized.
// ---------------------------------------------------------------------------
__global__ __launch_bounds__(256) void k_gru_dec(
    const float* __restrict__ enc,
    const float* __restrict__ wih0, const float* __restrict__ whh0,
    const float* __restrict__ bih0, const float* __restrict__ bhh0,
    const float* __restrict__ wih1, const float* __restrict__ whh1,
    const float* __restrict__ bih1, const float* __restrict__ bhh1,
    const float* __restrict__ dw1, const float* __restrict__ db1,
    const float* __restrict__ dw21, const float* __restrict__ db21,
    const float* __restrict__ dw23, const float* __restrict__ db23,
    float* __restrict__ out) {
  __shared__ float smem[8 * 1280];  // per wave: h0(16x32) h1(16x32) hd(16x16)
  const int tid = threadIdx.x;
  const int wave = tid >> 5, lane = tid & 31;
  const int lo = lane & 15, hi = lane >> 4;
  float* sh0 = smem + wave * 1280;
  float* sh1 = sh0 + 512;
  float* shd = sh1 + 512;

  for (int i = lane; i < 1024; i += 32) sh0[i] = 0.0f;  // h0 = h1 = 0
  wave_lds_sync();

  const int gw = blockIdx.x * 8 + wave;
  const int r0 = gw * 16;                // 16 consecutive rows (never cross b)
  const int b = r0 / Nn;
  const int n0 = r0 - b * Nn;

  v8f h0reg[2], h1reg[2];
#pragma unroll
  for (int j = 0; j < 2; ++j) { h0reg[j] = splat8(0.0f); h1reg[j] = splat8(0.0f); }

  for (int t = 0; t < Ww; ++t) {
    // ---------------- GRU layer 0 (input dim 64) ----------------
    v8f rz[4], gin[2], ghn[2];
#pragma unroll
    for (int j = 0; j < 4; ++j) rz[j] = splat8(bih0[16 * j + lo] + bhh0[16 * j + lo]);
#pragma unroll
    for (int j = 0; j < 2; ++j) {
      gin[j] = splat8(bih0[64 + 16 * j + lo]);
      ghn[j] = splat8(bhh0[64 + 16 * j + lo]);
    }
    const size_t xbase = ((size_t)(b * Ww + t) * Nn + n0 + lo) * Hh;
#pragma unroll 4
    for (int k = 0; k < 16; ++k) {  // K = 64
      const int kk = 4 * k + 2 * hi;
      v2f a = *reinterpret_cast<const v2f*>(enc + xbase + kk);
#pragma unroll
      for (int jj = 0; jj < 6; ++jj) {
        v2f bb = *reinterpret_cast<const v2f*>(wih0 + (size_t)(16 * jj + lo) * Hh + kk);
        if (jj < 4) rz[jj] = wmma4(a, bb, rz[jj]);
        else        gin[jj - 4] = wmma4(a, bb, gin[jj - 4]);
      }
    }
#pragma unroll
    for (int k = 0; k < 8; ++k) {  // K = 32 (hidden)
      const int kk = 4 * k + 2 * hi;
      v2f a = *reinterpret_cast<const v2f*>(sh0 + lo * Gg + kk);
#pragma unroll
      for (int jj = 0; jj < 6; ++jj) {
        v2f bb = *reinterpret_cast<const v2f*>(whh0 + (size_t)(16 * jj + lo) * Gg + kk);
        if (jj < 4) rz[jj] = wmma4(a, bb, rz[jj]);
        else        ghn[jj - 4] = wmma4(a, bb, ghn[jj - 4]);
      }
    }
#pragma unroll
    for (int j = 0; j < 2; ++j)
#pragma unroll
      for (int p = 0; p < 8; ++p) {
        float r = sigmoidf_(rz[j][p]);
        float z = sigmoidf_(rz[2 + j][p]);
        float nn = tanhf(gin[j][p] + r * ghn[j][p]);
        float hn = (1.0f - z) * nn + z * h0reg[j][p];
        h0reg[j][p] = hn;
        sh0[(p + 8 * hi) * Gg + 16 * j + lo] = hn;
      }
    wave_lds_sync();

    // ---------------- GRU layer 1 (input dim 32) ----------------
    v8f rz1[4], gin1[2], ghn1[2];
#pragma unroll
    for (int j = 0; j < 4; ++j) rz1[j] = splat8(bih1[16 * j + lo] + bhh1[16 * j + lo]);
#pragma unroll
    for (int j = 0; j < 2; ++j) {
      gin1[j] = splat8(bih1[64 + 16 * j + lo]);
      ghn1[j] = splat8(bhh1[64 + 16 * j + lo]);
    }
#pragma unroll
    for (int k = 0; k < 8; ++k) {  // x = h0_new from LDS
      const int kk = 4 * k + 2 * hi;
      v2f a = *reinterpret_cast<const v2f*>(sh0 + lo * Gg + kk);
#pragma unroll
      for (int jj = 0; jj < 6; ++jj) {
        v2f bb = *reinterpret_cast<const v2f*>(wih1 + (size_t)(16 * jj + lo) * Gg + kk);
        if (jj < 4) rz1[jj] = wmma4(a, bb, rz1[jj]);
        else        gin1[jj - 4] = wmma4(a, bb, gin1[jj - 4]);
      }
    }
#pragma unroll
    for (int k = 0; k < 8; ++k) {  // h1_old from LDS
      const int kk = 4 * k + 2 * hi;
      v2f a = *reinterpret_cast<const v2f*>(sh1 + lo * Gg + kk);
#pragma unroll
      for (int jj = 0; jj < 6; ++jj) {
        v2f bb = *reinterpret_cast<const v2f*>(whh1 + (size_t)(16 * jj + lo) * Gg + kk);
        if (jj < 4) rz1[jj] = wmma4(a, bb, rz1[jj]);
        else        ghn1[jj - 4] = wmma4(a, bb, ghn1[jj - 4]);
      }
    }
#pragma unroll
    for (int j = 0; j < 2; ++j)
#pragma unroll
      for (int p = 0; p < 8; ++p) {
        float r = sigmoidf_(rz1[j][p]);
        float z = sigmoidf_(rz1[2 + j][p]);
        float nn = tanhf(gin1[j][p] + r * ghn1[j][p]);
        float hn = (1.0f - z) * nn + z * h1reg[j][p];
        h1reg[j][p] = hn;
        sh1[(p + 8 * hi) * Gg + 16 * j + lo] = hn;
      }
    wave_lds_sync();

    // ---------------- decoder: 6 heads, z = h1_new (16x32) ----------------
#pragma unroll
    for (int k6 = 0; k6 < 6; ++k6) {
      v8f acc = splat8(db1[k6 * 16 + lo]);
#pragma unroll
      for (int s = 0; s < 8; ++s) {  // K = 32
        const int kk = 4 * s + 2 * hi;
        v2f a = *reinterpret_cast<const v2f*>(sh1 + lo * Gg + kk);
        v2f bb;
        bb.x = dw1[((size_t)k6 * Gg + kk) * 16 + lo];
        bb.y = dw1[((size_t)k6 * Gg + kk + 1) * 16 + lo];
        acc = wmma4(a, bb, acc);
      }
#pragma unroll
      for (int p = 0; p < 8; ++p)
        shd[(p + 8 * hi) * 16 + lo] = fmaxf(acc[p], 0.0f);
      wave_lds_sync();

      const int Ok = (k6 < 3) ? 1 : 3;
      const int cb = (k6 < 3) ? k6 : (3 + (k6 - 3) * 3);
      for (int v = lane; v < 16 * Ok; v += 32) {
        const int m = v & 15, o = v >> 4;
        float s;
        if (k6 < 3) {
          s = db21[k6];
          const float* w2 = dw21 + k6 * 16;
#pragma unroll
          for (int d = 0; d < 16; ++d) s += shd[m * 16 + d] * w2[d];
        } else {
          const int kq = k6 - 3;
          s = db23[kq * 3 + o];
          const float* w2 = dw23 + (size_t)kq * 16 * 3;
#pragma unroll
          for (int d = 0; d < 16; ++d) s += shd[m * 16 + d] * w2[d * 3 + o];
        }
        out[((size_t)(r0 + m) * Ww + t) * 12 + cb + o] = s;
      }
      wave_lds_sync();
    }
  }
}

// ---------------------------------------------------------------------------
// Host launcher
// ---------------------------------------------------------------------------
extern "C" void kernel_launch(void* const* d_in, const int* in_sizes, int n_in,
                              void* d_out, int out_size, void* d_ws, size_t ws_size,
                              hipStream_t stream) {
  (void)in_sizes; (void)n_in; (void)out_size; (void)ws_size;
  const float* x    = (const float*)d_in[0];
  const int*   ei   = (const int*)d_in[1];
  const float* pW   = (const float*)d_in[2];
  const float* pb   = (const float*)d_in[3];
  const float* gW   = (const float*)d_in[4];
  const float* gb   = (const float*)d_in[5];
  const float* wih0 = (const float*)d_in[6];
  const float* whh0 = (const float*)d_in[7];
  const float* bih0 = (const float*)d_in[8];
  const float* bhh0 = (const float*)d_in[9];
  const float* wih1 = (const float*)d_in[10];
  const float* whh1 = (const float*)d_in[11];
  const float* bih1 = (const float*)d_in[12];
  const float* bhh1 = (const float*)d_in[13];
  const float* dw1  = (const float*)d_in[14];
  const float* db1  = (const float*)d_in[15];
  const float* dw21 = (const float*)d_in[16];
  const float* db21 = (const float*)d_in[17];
  const float* dw23 = (const float*)d_in[18];
  const float* db23 = (const float*)d_in[19];
  float* out = (float*)d_out;

  char* ws = (char*)d_ws;
  const size_t encB = (size_t)BT * Nn * Hh * sizeof(float);  // ~393 MB
  float* bufA = (float*)ws;                    // activations (ping)
  float* bufB = (float*)(ws + encB);           // activations (pong)
  float* dis  = (float*)(ws + 2 * encB);       // N floats (deg -> rsqrt(deg))
  float* nw   = dis + Nn;                      // E floats (edge norm)
  const int* srcI = ei;
  const int* dstI = ei + Ee;

  k_deg_init<<<(Nn + 255) / 256, 256, 0, stream>>>(dis);
  k_deg_count<<<(Ee + 255) / 256, 256, 0, stream>>>(dstI, dis);
  k_rsqrt_inplace<<<(Nn + 255) / 256, 256, 0, stream>>>(dis);
  k_edgenorm<<<(Ee + 255) / 256, 256, 0, stream>>>(srcI, dstI, dis, nw);

  k_project<<<TILES / 4, 128, 0, stream>>>(x, pW, pb, bufA);

  const int EW = BT * Nn * Hh / 256;   // 384000 blocks, exact
  const int SC = BT * Ee * 16 / 256;   // 384000 blocks, exact
  for (int l = 0; l < Ll; ++l) {
    k_gcn_mm<<<TILES / 4, 128, 0, stream>>>(bufA, gW + (size_t)l * Hh * Hh, bufB);
    k_selfinit<<<EW, 256, 0, stream>>>(bufB, dis, bufA);
    k_scatter<<<SC, 256, 0, stream>>>(bufB, srcI, dstI, nw, bufA);
    k_biasrelu<<<EW, 256, 0, stream>>>(bufA, gb + l * Hh);
  }

  k_gru_dec<<<ROWS / 16 / 8, 256, 0, stream>>>(bufA, wih0, whh0, bih0, bhh0,
                                               wih1, whh1, bih1, bhh1,
                                               dw1, db1, dw21, db21, dw23, db23,
                                               out);
}